// ScaledDotProdAttV1Pool1d_7928509628952
// MI455X (gfx1250) — compile-verified
//
#include <hip/hip_runtime.h>

// ---------------------------------------------------------------------------
// Problem constants (from reference)
// ---------------------------------------------------------------------------
#define BB   32
#define FF   512
#define TTOT 4000
#define HH   8
#define DKV  64            // DK == DV == 64
#define NS   25            // T splits  (25 * 10 tiles * 16 = 4000)
#define TILES 10
#define TT   16            // t-tile (WMMA N for scores, K for y-GEMM)
#define XP   20            // LDS pitch for x tile   (even, 20l mod 64 distinct l=0..15)
#define QP   516           // LDS pitch for qkw      (bank = (4h+f) mod 64)
#define PP   20            // LDS pitch for p tile
#define NEGF (-1.0e38f)

#define GLOBAL_AS __attribute__((address_space(1)))
#define LDS_AS    __attribute__((address_space(3)))

#if __has_builtin(__builtin_amdgcn_global_load_async_to_lds_b128)
#define HAVE_ASYNC_LDS 1
#else
#define HAVE_ASYNC_LDS 0
#endif

typedef float v2f __attribute__((ext_vector_type(2)));
typedef float v8f __attribute__((ext_vector_type(8)));
typedef int   v4i __attribute__((ext_vector_type(4)));

__device__ __forceinline__ v8f wmma_f32(v2f a, v2f b, v8f c) {
  // D = A(16x4 f32) * B(4x16 f32) + C(16x16 f32)
  return __builtin_amdgcn_wmma_f32_16x16x4_f32(false, a, false, b,
                                               (short)0, c, false, false);
}

// ---------------------------------------------------------------------------
// Kernel 1: fold q into Wk:  qkw[h,f] = (1/8) * sum_d q[h,d]*Wk[h*64+d, f]
//           qb[h] = (1/8) * sum_d q[h,d]*bk[h*64+d]       (1/sqrt(64) = 1/8)
// ---------------------------------------------------------------------------
__global__ __launch_bounds__(256) void prep_kernel(
    const float* __restrict__ q, const float* __restrict__ Wk,
    const float* __restrict__ bk, float* __restrict__ qkw,
    float* __restrict__ qb) {
  const int h = blockIdx.x;
  const int tid = threadIdx.x;
  for (int f = tid; f < FF; f += 256) {
    float a = 0.f;
    #pragma unroll 8
    for (int d = 0; d < DKV; ++d)
      a += q[h * DKV + d] * Wk[(size_t)(h * DKV + d) * FF + f];
    qkw[h * FF + f] = a * 0.125f;
  }
  if (tid == 0) {
    float a = 0.f;
    for (int d = 0; d < DKV; ++d) a += q[h * DKV + d] * bk[h * DKV + d];
    qb[h] = a * 0.125f;
  }
}

// ---------------------------------------------------------------------------
// Kernel 2: single pass over x, flash-style.
// Block (split, b): 8 wave32s. Per t-tile of 16:
//   * async-copy x[b, :, t0:t0+16] -> LDS (512 x 16, pitch 20)
//   * scores s[h,t] = qkw @ xtile   : WMMA f32 16x16x4, K=512 split on waves 0-3
//   * online softmax update per head (wave h owns head h)
//   * y[h,f] += p[h,t] * x[f,t]     : WMMA, 8 waves x 64 f-columns, K=16
// Emits per-(b,split) partials: m[h], l[h], y[8][512] (unnormalized).
// ---------------------------------------------------------------------------
__global__ __launch_bounds__(256) void flash_kernel(
    const float* __restrict__ x, const int* __restrict__ wmask,
    const float* __restrict__ qkw, const float* __restrict__ qb,
    float* __restrict__ y_part, float* __restrict__ ml_part) {
  __shared__ float xs[FF * XP];          // 40960 B
  __shared__ float qs[HH * QP];          // 16512 B
  __shared__ float ps[HH * PP];          //   640 B
  __shared__ float sred[4][HH][TT];      //  2048 B
  __shared__ float fac_s[HH];

  const int tid   = threadIdx.x;
  const int lane  = tid & 31;
  const int wv    = tid >> 5;            // wave 0..7 == head id in flash phase
  const int split = blockIdx.x;          // 0..NS-1
  const int b     = blockIdx.y;          // 0..31

  // Stage qkw into LDS (banked pitch).
  for (int i = tid; i < HH * FF; i += 256) {
    int h = i >> 9, f = i & (FF - 1);
    qs[h * QP + f] = qkw[i];
  }
  const float qb_h = qb[wv];

  // WMMA lane decomposition (wave32): lanes 0-15 = rows/cols 0-15 of first
  // K-half, lanes 16-31 carry K+2 (per ISA 16x4 / 4x16 f32 layouts).
  const int rA   = lane & 15;
  const int half = lane >> 4;
  const int ko   = half << 1;            // K offset 0 or 2
  const int r8   = rA & 7;               // A rows 8-15 duplicate heads (discarded)

  v8f acc0 = {0.f,0.f,0.f,0.f,0.f,0.f,0.f,0.f};
  v8f acc1 = acc0, acc2 = acc0, acc3 = acc0;
  float m_run = NEGF, l_run = 0.f;

  const float* xb = x + (size_t)b * FF * TTOT;
  const int t_begin = split * (TT * TILES);

  for (int tile = 0; tile < TILES; ++tile) {
    const int t0 = t_begin + tile * TT;
    __syncthreads();                               // previous tile fully consumed

    // ---- stage x tile: 512 rows x 16 floats (b128 chunks) -----------------
#if HAVE_ASYNC_LDS
    // CDNA5 async global->LDS DMA path: no VGPR staging, tracked by ASYNCcnt.
    #pragma unroll
    for (int i2 = 0; i2 < 8; ++i2) {
      const int i = (i2 << 8) + tid;
      const int f = i >> 2, c = (i & 3) << 2;
      GLOBAL_AS v4i* gp = (GLOBAL_AS v4i*)(xb + (size_t)f * TTOT + t0 + c);
      LDS_AS    v4i* lp = (LDS_AS v4i*)&xs[f * XP + c];
      __builtin_amdgcn_global_load_async_to_lds_b128(gp, lp, 0, 0);
    }
#if __has_builtin(__builtin_amdgcn_s_wait_asynccnt)
    __builtin_amdgcn_s_wait_asynccnt(0);
#else
    asm volatile("s_wait_asynccnt 0x0" ::: "memory");
#endif
#else
    // Fallback: software-pipelined copy (all loads in flight, then stores).
    float4 stage[8];
    #pragma unroll
    for (int i2 = 0; i2 < 8; ++i2) {
      const int i = (i2 << 8) + tid;
      const int f = i >> 2, c = (i & 3) << 2;
      stage[i2] = *(const float4*)(xb + (size_t)f * TTOT + t0 + c);
    }
    #pragma unroll
    for (int i2 = 0; i2 < 8; ++i2) {
      const int i = (i2 << 8) + tid;
      const int f = i >> 2, c = (i & 3) << 2;
      *(float4*)&xs[f * XP + c] = stage[i2];       // XP%4==0 -> 16B aligned
    }
#endif
    if (tile + 1 < TILES) {                        // prefetch next tile
      const float* nx = xb + (size_t)tid * TTOT + t0 + TT;
      __builtin_prefetch(nx, 0, 3);
      __builtin_prefetch(nx + (size_t)256 * TTOT, 0, 3);
    }
    __syncthreads();

    // ---- scores: waves 0-3, K slices of 128 (f) ---------------------------
    if (wv < 4) {
      v8f sc = {0.f,0.f,0.f,0.f,0.f,0.f,0.f,0.f};
      const int fk0 = wv << 7;
      #pragma unroll
      for (int kk = 0; kk < 128; kk += 4) {
        const int fk = fk0 + kk;
        v2f a = *(const v2f*)&qs[r8 * QP + fk + ko];          // A: qkw 16x4
        v2f bb;                                               // B: x 4x16 (col=t)
        bb.x = xs[(fk + ko) * XP + rA];
        bb.y = xs[(fk + ko + 1) * XP + rA];
        sc = wmma_f32(a, bb, sc);
      }
      if (half == 0) {                  // rows 0-7 live in lanes 0-15, vgpr r
        #pragma unroll
        for (int r = 0; r < 8; ++r) sred[wv][r][rA] = sc[r];
      }
    }
    __syncthreads();

    // ---- online softmax update: wave wv handles head wv -------------------
    {
      float s = NEGF;
      bool valid = false;
      if (lane < TT) {
        float sv = sred[0][wv][lane] + sred[1][wv][lane] +
                   sred[2][wv][lane] + sred[3][wv][lane] + qb_h;
        if (wmask[(size_t)b * TTOT + t0 + lane] != 0) { s = sv; valid = true; }
      }
      float tm = s;
      for (int off = 16; off; off >>= 1) tm = fmaxf(tm, __shfl_xor(tm, off, 32));
      const float nm  = fmaxf(m_run, tm);
      const float fac = __expf(m_run - nm);        // nm==m_run==NEGF -> exp(0)=1
      float p = valid ? __expf(s - nm) : 0.f;
      float psum = p;
      for (int off = 16; off; off >>= 1) psum += __shfl_xor(psum, off, 32);
      l_run = l_run * fac + psum;
      m_run = nm;
      if (lane < TT)  ps[wv * PP + lane] = p;
      if (lane == 0)  fac_s[wv] = fac;
    }
    __syncthreads();

    // ---- rescale accumulators, then y-GEMM (all 8 waves) ------------------
    {
      float fr[8];
      #pragma unroll
      for (int r = 0; r < 8; ++r) fr[r] = fac_s[r];
      #pragma unroll
      for (int r = 0; r < 8; ++r) {
        acc0[r] *= fr[r]; acc1[r] *= fr[r]; acc2[r] *= fr[r]; acc3[r] *= fr[r];
      }
      const int fb = wv << 6;                       // 64 f-columns per wave
      #pragma unroll
      for (int ks = 0; ks < 4; ++ks) {
        const int tk = ks << 2;
        v2f a  = *(const v2f*)&ps[r8 * PP + tk + ko];               // A: p 16x4
        v2f b0 = *(const v2f*)&xs[(fb +  0 + rA) * XP + tk + ko];   // B: x^T 4x16
        v2f b1 = *(const v2f*)&xs[(fb + 16 + rA) * XP + tk + ko];
        v2f b2 = *(const v2f*)&xs[(fb + 32 + rA) * XP + tk + ko];
        v2f b3 = *(const v2f*)&xs[(fb + 48 + rA) * XP + tk + ko];
        acc0 = wmma_f32(a, b0, acc0);
        acc1 = wmma_f32(a, b1, acc1);
        acc2 = wmma_f32(a, b2, acc2);
        acc3 = wmma_f32(a, b3, acc3);
      }
    }
  }

  // ---- emit partials ------------------------------------------------------
  if (lane < 16) {
    float* yo = y_part + (size_t)(b * NS + split) * HH * FF;
    const int f0 = (wv << 6) + lane;
    #pragma unroll
    for (int r = 0; r < 8; ++r) {
      yo[(size_t)r * FF + f0     ] = acc0[r];
      yo[(size_t)r * FF + f0 + 16] = acc1[r];
      yo[(size_t)r * FF + f0 + 32] = acc2[r];
      yo[(size_t)r * FF + f0 + 48] = acc3[r];
    }
  }
  if (lane == 0) {
    ml_part[(size_t)(b * NS + split) * 16 + wv]      = m_run;
    ml_part[(size_t)(b * NS + split) * 16 + 8 + wv]  = l_run;
  }
}

// ---------------------------------------------------------------------------
// Kernel 3: combine split partials + project through Wv.
//   out[b, h*64+dv] = Wv_h @ (Y[h]/L[h]) + bv  (0 if row fully masked)
// ---------------------------------------------------------------------------
__global__ __launch_bounds__(256) void combine_kernel(
    const float* __restrict__ y_part, const float* __restrict__ ml_part,
    const float* __restrict__ Wv, const float* __restrict__ bv,
    float* __restrict__ out) {
  __shared__ float Ys[HH * FF];          // 16 KB
  __shared__ float facs[NS * HH];
  __shared__ float Ls[HH];

  const int b = blockIdx.x;
  const int tid = threadIdx.x, lane = tid & 31, wv = tid >> 5;

  // Stage A: per head (wave wv): global max + weighted denominator.
  {
    float m = NEGF, l = 0.f;
    if (lane < NS) {
      m = ml_part[(size_t)(b * NS + lane) * 16 + wv];
      l = ml_part[(size_t)(b * NS + lane) * 16 + 8 + wv];
    }
    float M = m;
    for (int off = 16; off; off >>= 1) M = fmaxf(M, __shfl_xor(M, off, 32));
    float fac = (lane < NS) ? __expf(m - M) : 0.f;
    float L = l * fac;
    for (int off = 16; off; off >>= 1) L += __shfl_xor(L, off, 32);
    if (lane < NS) facs[lane * HH + wv] = fac;
    if (lane == 0) Ls[wv] = L;
  }
  __syncthreads();

  // Stage B: Y[h,f] = sum_s y_s * fac_s / L
  for (int idx = tid; idx < HH * FF; idx += 256) {
    const int h = idx >> 9, f = idx & (FF - 1);
    float a = 0.f;
    #pragma unroll 5
    for (int s = 0; s < NS; ++s)
      a += y_part[((size_t)(b * NS + s) * HH + h) * FF + f] * facs[s * HH + h];
    const float L = Ls[h];
    Ys[idx] = (L > 0.f) ? a / L : 0.f;
  }
  __syncthreads();

  // Stage C: projection. Each wave computes 64 outputs cooperatively
  // (coalesced Wv reads, conflict-free LDS reads, wave-sum reduce).
  for (int j = 0; j < 64; ++j) {
    const int c = wv + (j << 3);          // 0..511
    const int h = c >> 6;
    float a = 0.f;
    #pragma unroll
    for (int k = 0; k < 16; ++k)
      a += Wv[(size_t)c * FF + lane + (k << 5)] * Ys[h * FF + lane + (k << 5)];
    for (int off = 16; off; off >>= 1) a += __shfl_xor(a, off, 32);
    if (lane == 0)
      out[(size_t)b * (HH * DKV) + c] = a + ((Ls[h] > 0.f) ? bv[c] : 0.f);
  }
}

// ---------------------------------------------------------------------------
extern "C" void kernel_launch(void* const* d_in, const int* in_sizes, int n_in,
                              void* d_out, int out_size, void* d_ws, size_t ws_size,
                              hipStream_t stream) {
  const float* x  = (const float*)d_in[0];
  const int*   w  = (const int*)  d_in[1];
  const float* q  = (const float*)d_in[2];
  const float* Wk = (const float*)d_in[3];
  const float* bk = (const float*)d_in[4];
  const float* Wv = (const float*)d_in[5];
  const float* bv = (const float*)d_in[6];
  float* out = (float*)d_out;

  float* ws  = (float*)d_ws;
  float* qkw = ws;                        // 8*512            = 4096
  float* qb  = ws + 4096;                 // 8 (padded to 16)
  float* ml  = ws + 4112;                 // 32*25*16         = 12800
  float* yp  = ws + 16912;                // 32*25*8*512      = 3,276,800 (~13.1 MB)

  prep_kernel<<<HH, 256, 0, stream>>>(q, Wk, bk, qkw, qb);
  flash_kernel<<<dim3(NS, BB), 256, 0, stream>>>(x, w, qkw, qb, yp, ml);
  combine_kernel<<<BB, 256, 0, stream>>>(yp, ml, Wv, bv, out);
}